// CombinedHiddenPRADADecoder_369367188152
// MI455X (gfx1250) — compile-verified
//
#include <hip/hip_runtime.h>
#include <hip/hip_bf16.h>

typedef float v2f __attribute__((ext_vector_type(2)));
typedef float v8f __attribute__((ext_vector_type(8)));

#define NNODES 50000

// ---------------------------------------------------------------------------
// Degree / normalization kernels
// ---------------------------------------------------------------------------
__global__ void k_init_deg(float* deg) {
  int i = blockIdx.x * blockDim.x + threadIdx.x;
  if (i < NNODES) deg[i] = 1.0f;  // self loop
}

__global__ void k_count_deg(const int* __restrict__ dst, float* __restrict__ deg, int E) {
  int e = blockIdx.x * blockDim.x + threadIdx.x;
  if (e < E) atomicAdd(&deg[dst[e]], 1.0f);
}

__global__ void k_finish_deg(float* deg) {
  int i = blockIdx.x * blockDim.x + threadIdx.x;
  if (i < NNODES) deg[i] = rsqrtf(deg[i]);  // deg >= 1 always
}

// ---------------------------------------------------------------------------
// fp32 WMMA GEMM: C[M,N] = A[M,K] * W[K,N]; M%16==0, N%16==0, K%4==0.
// One wave per 16x16 output tile, V_WMMA_F32_16X16X4_F32 along K.
// ---------------------------------------------------------------------------
template <int K, int N>
__global__ void k_gemm_wmma_f32(const float* __restrict__ A,
                                const float* __restrict__ W,
                                float* __restrict__ C, int M) {
  const int wave = (int)((blockIdx.x * blockDim.x + threadIdx.x) >> 5);
  const int lane = threadIdx.x & 31;
  const int tiles_n = N / 16;
  const int row_tile = wave / tiles_n;
  const int col_tile = wave % tiles_n;
  if (row_tile * 16 >= M) return;  // wave-uniform

  const int half = lane >> 4;   // 0: K pair {k0,k0+1}; 1: {k0+2,k0+3}
  const int r    = lane & 15;   // A: row M; B: col N

  const float* arow = A + (size_t)(row_tile * 16 + r) * K;   // A[m][*]
  const float* wcol = W + (size_t)(col_tile * 16 + r);       // W[*][n]

  v8f acc = {};
#pragma unroll 4
  for (int k0 = 0; k0 < K; k0 += 4) {
    const int kA = k0 + 2 * half;
    v2f a = *(const v2f*)(arow + kA);          // A[m][kA], A[m][kA+1]
    v2f b;
    b.x = wcol[(size_t)kA * N];                // W[kA][n]
    b.y = wcol[(size_t)(kA + 1) * N];          // W[kA+1][n]
    acc = __builtin_amdgcn_wmma_f32_16x16x4_f32(
        /*neg_a=*/false, a, /*neg_b=*/false, b,
        /*c_mod=*/(short)0, acc, /*reuse_a=*/false, /*reuse_b=*/false);
  }

  // D layout: VGPR i, lanes 0-15 -> M=i, lanes 16-31 -> M=i+8; N = lane&15
  float* crow = C + (size_t)(row_tile * 16 + 8 * half) * N + col_tile * 16 + r;
#pragma unroll
  for (int i = 0; i < 8; ++i) crow[(size_t)i * N] = acc[i];
}

// ---------------------------------------------------------------------------
// Tile init: out[n][c] = feat[n][c]*dinv[n]^2 + bias[c]   (self loop + bias)
// ---------------------------------------------------------------------------
template <int CH>
__global__ void k_self_bias(const float* __restrict__ feat,
                            const float* __restrict__ dinv,
                            const float* __restrict__ bias,
                            float* __restrict__ out) {
  int idx = blockIdx.x * blockDim.x + threadIdx.x;
  if (idx >= NNODES * CH) return;
  int n = idx / CH;
  int c = idx % CH;
  float di = dinv[n];
  out[idx] = feat[idx] * di * di + bias[c];
}

// ---------------------------------------------------------------------------
// Edge scatter: out[dst] += feat[src] * dinv[src]*dinv[dst].
// One wave per edge; lane covers CH/32 consecutive channels.
// ---------------------------------------------------------------------------
template <int CH>
__global__ void k_scatter(const int* __restrict__ src, const int* __restrict__ dst,
                          const float* __restrict__ dinv,
                          const float* __restrict__ feat,
                          float* __restrict__ out, int E) {
  const int wave = (int)((blockIdx.x * blockDim.x + threadIdx.x) >> 5);
  const int lane = threadIdx.x & 31;
  if (wave >= E) return;
  const int s = src[wave];
  const int d = dst[wave];
  const float w = dinv[s] * dinv[d];
  const float* frow = feat + (size_t)s * CH;
  float* orow = out + (size_t)d * CH;
  if (CH == 128) {
    float4 v = *(const float4*)(frow + lane * 4);
    atomicAdd(orow + lane * 4 + 0, v.x * w);
    atomicAdd(orow + lane * 4 + 1, v.y * w);
    atomicAdd(orow + lane * 4 + 2, v.z * w);
    atomicAdd(orow + lane * 4 + 3, v.w * w);
  } else {  // CH == 64
    float2 v = *(const float2*)(frow + lane * 2);
    atomicAdd(orow + lane * 2 + 0, v.x * w);
    atomicAdd(orow + lane * 2 + 1, v.y * w);
  }
}

// ---------------------------------------------------------------------------
// Elementwise tanh
// ---------------------------------------------------------------------------
__global__ void k_tanh(const float* __restrict__ in, float* __restrict__ out, int n) {
  int idx = blockIdx.x * blockDim.x + threadIdx.x;
  if (idx < n) out[idx] = tanhf(in[idx]);
}

// ---------------------------------------------------------------------------
// Launch
// ---------------------------------------------------------------------------
extern "C" void kernel_launch(void* const* d_in, const int* in_sizes, int n_in,
                              void* d_out, int out_size, void* d_ws, size_t ws_size,
                              hipStream_t stream) {
  const float* x   = (const float*)d_in[0];   // [50000,256]
  const int*  eidx = (const int*)d_in[1];     // [2,E]
  const float* W1  = (const float*)d_in[2];   // [256,128]
  const float* b1  = (const float*)d_in[3];   // [128]
  const float* W2  = (const float*)d_in[4];   // [128,64]
  const float* b2  = (const float*)d_in[5];   // [64]
  float* out = (float*)d_out;                 // [50000,64]

  const int E = in_sizes[1] / 2;
  const int* src = eidx;
  const int* dst = eidx + E;

  // Workspace layout (floats): dinv | bufA (50000*128) | bufB (50000*128)
  float* dinv = (float*)d_ws;
  float* bufA = dinv + 50048;            // xw, then h
  float* bufB = bufA + (size_t)NNODES * 128;  // pre-act layer1, then hw

  const int T = 256;

  // --- degree / dinv ---
  k_init_deg<<<(NNODES + T - 1) / T, T, 0, stream>>>(dinv);
  k_count_deg<<<(E + T - 1) / T, T, 0, stream>>>(dst, dinv, E);
  k_finish_deg<<<(NNODES + T - 1) / T, T, 0, stream>>>(dinv);

  // --- layer 1: xw = x @ W1 ---
  {
    int waves = (NNODES / 16) * (128 / 16);      // 25000
    k_gemm_wmma_f32<256, 128><<<waves * 32 / 128, 128, 0, stream>>>(x, W1, bufA, NNODES);
  }
  k_self_bias<128><<<(NNODES * 128 + T - 1) / T, T, 0, stream>>>(bufA, dinv, b1, bufB);
  k_scatter<128><<<(E + 7) / 8, 256, 0, stream>>>(src, dst, dinv, bufA, bufB, E);
  k_tanh<<<(NNODES * 128 + T - 1) / T, T, 0, stream>>>(bufB, bufA, NNODES * 128);

  // --- layer 2: hw = h @ W2 ---
  {
    int waves = (NNODES / 16) * (64 / 16);       // 12500
    k_gemm_wmma_f32<128, 64><<<waves * 32 / 128, 128, 0, stream>>>(bufA, W2, bufB, NNODES);
  }
  k_self_bias<64><<<(NNODES * 64 + T - 1) / T, T, 0, stream>>>(bufB, dinv, b2, out);
  k_scatter<64><<<(E + 7) / 8, 256, 0, stream>>>(src, dst, dinv, bufB, out, E);
}